// AttnBYOL_8108898255038
// MI455X (gfx1250) — compile-verified
//
#include <hip/hip_runtime.h>
#include <hip/hip_bf16.h>

typedef float v2f __attribute__((ext_vector_type(2)));
typedef float v8f __attribute__((ext_vector_type(8)));

namespace cfg {
constexpr int B = 32, L = 1024, H = 3, F = 81, D = 243, NF = 243, FFH = 972;
constexpr long long BL   = (long long)B * L;              // 32768
constexpr long long BLD  = BL * D;                        // 7,962,624
constexpr long long HBL  = (long long)H * BL;             // 98,304
constexpr long long PEL  = BL * FFH;                      // 31,850,496 (>= H*BL*NF = 23,887,872)
constexpr long long CTXE = (long long)H * B * NF * F;     // 1,889,568
constexpr long long KSUME = (long long)H * B * NF;        // 23,328
constexpr float DN    = 1.0f / 3.0f;                      // F^-0.25, F=81
constexpr float RATIO = 0.06415002990995843f;             // NF^-0.5, NF=243
}

static __device__ __forceinline__ long long llmin(long long a, long long b) { return a < b ? a : b; }
static __device__ __forceinline__ int imin(int a, int b) { return a < b ? a : b; }

// ---------------------------------------------------------------------------
// Generic strided batched GEMM, fp32 WMMA 16x16x4.
// C[m,n] (+epilogue) = sum_k A[offA + m*sAm + k*sAk] * B[offB + k*sBk + n*sBn]
// Per-batch offsets: off = (z % bdiv)*lo + (z / bdiv)*hi   (z = blockIdx.z)
// Block = 128 threads = 4 waves; wave computes 16 rows x 64 cols (4 subtiles).
// Out-of-range m/n lanes read CLAMPED (valid) addresses; their garbage only
// lands in C rows/cols that the guarded store never writes. The K tail is the
// only place that needs value masking, and it is branch-free (load + select).
// EPI: 0 = val*alpha; 1 = val+bias[n]; 2 = gelu(val+bias[n]); 3 = val*rs[row]
// ---------------------------------------------------------------------------
template <int EPI>
__global__ __launch_bounds__(128) void k_gemm(
    const float* __restrict__ A, long long sAm, long long sAk, long long aLo, long long aHi,
    const float* __restrict__ Bm, long long sBk, long long sBn, long long bLo, long long bHi,
    float* __restrict__ C, long long ldc, long long cLo, long long cHi,
    int M, int N, int K, int bdiv, float alpha,
    const float* __restrict__ bias, const float* __restrict__ rs,
    long long rLo, long long rHi)
{
    const int z  = blockIdx.z;
    const long long zm = z % bdiv, zd = z / bdiv;
    const long long offA = zm * aLo + zd * aHi;
    const long long offB = zm * bLo + zd * bHi;
    const long long offC = zm * cLo + zd * cHi;
    const long long offR = zm * rLo + zd * rHi;

    const int lane  = threadIdx.x & 31;
    const int wave  = threadIdx.x >> 5;
    const int hi    = lane >> 4;        // 0: lanes 0-15, 1: lanes 16-31
    const int nl    = lane & 15;
    const int mBase = blockIdx.y * 64 + wave * 16;
    const int nBase = blockIdx.x * 64;

    // clamped per-lane coordinates (always-valid addresses, see header comment)
    const int mA = imin(mBase + nl, M - 1);
    const int n0 = imin(nBase +  0 + nl, N - 1);
    const int n1 = imin(nBase + 16 + nl, N - 1);
    const int n2 = imin(nBase + 32 + nl, N - 1);
    const int n3 = imin(nBase + 48 + nl, N - 1);

    const float* __restrict__ pA = A + offA + (long long)mA * sAm + (long long)(2 * hi) * sAk;
    const float* __restrict__ bb0 = Bm + offB + (long long)(2 * hi) * sBk;
    const float* __restrict__ pB0 = bb0 + (long long)n0 * sBn;
    const float* __restrict__ pB1 = bb0 + (long long)n1 * sBn;
    const float* __restrict__ pB2 = bb0 + (long long)n2 * sBn;
    const float* __restrict__ pB3 = bb0 + (long long)n3 * sBn;
    const long long aStep = 4 * sAk;
    const long long bStep = 4 * sBk;

    v8f acc0 = {}, acc1 = {}, acc2 = {}, acc3 = {};

    const int K4 = K & ~3;              // full 4-wide steps: no k-guards needed
    for (int k0 = 0; k0 < K4; k0 += 4) {
        v2f av  = { pA[0],  pA[sAk]  };
        v2f bv0 = { pB0[0], pB0[sBk] };
        v2f bv1 = { pB1[0], pB1[sBk] };
        v2f bv2 = { pB2[0], pB2[sBk] };
        v2f bv3 = { pB3[0], pB3[sBk] };
        acc0 = __builtin_amdgcn_wmma_f32_16x16x4_f32(false, av, false, bv0, (short)0, acc0, false, false);
        acc1 = __builtin_amdgcn_wmma_f32_16x16x4_f32(false, av, false, bv1, (short)0, acc1, false, false);
        acc2 = __builtin_amdgcn_wmma_f32_16x16x4_f32(false, av, false, bv2, (short)0, acc2, false, false);
        acc3 = __builtin_amdgcn_wmma_f32_16x16x4_f32(false, av, false, bv3, (short)0, acc3, false, false);
        pA += aStep; pB0 += bStep; pB1 += bStep; pB2 += bStep; pB3 += bStep;
    }

    if (K4 < K) {                       // fractional K tail, branch-free masking
        const int kA = K4 + 2 * hi;
        const long long kc0 = llmin(kA,     K - 1) * sAk;
        const long long kc1 = llmin(kA + 1, K - 1) * sAk;
        const long long kb0 = llmin(kA,     K - 1) * sBk;
        const long long kb1 = llmin(kA + 1, K - 1) * sBk;
        const bool ok0 = kA     < K;
        const bool ok1 = kA + 1 < K;
        const float* qA = A + offA + (long long)mA * sAm;
        const float* qB = Bm + offB;
        float a0 = qA[kc0], a1 = qA[kc1];
        a0 = ok0 ? a0 : 0.0f;  a1 = ok1 ? a1 : 0.0f;
        v2f av = { a0, a1 };
        float u0, u1;
        u0 = qB[kb0 + (long long)n0 * sBn]; u1 = qB[kb1 + (long long)n0 * sBn];
        v2f bv0 = { ok0 ? u0 : 0.0f, ok1 ? u1 : 0.0f };
        u0 = qB[kb0 + (long long)n1 * sBn]; u1 = qB[kb1 + (long long)n1 * sBn];
        v2f bv1 = { ok0 ? u0 : 0.0f, ok1 ? u1 : 0.0f };
        u0 = qB[kb0 + (long long)n2 * sBn]; u1 = qB[kb1 + (long long)n2 * sBn];
        v2f bv2 = { ok0 ? u0 : 0.0f, ok1 ? u1 : 0.0f };
        u0 = qB[kb0 + (long long)n3 * sBn]; u1 = qB[kb1 + (long long)n3 * sBn];
        v2f bv3 = { ok0 ? u0 : 0.0f, ok1 ? u1 : 0.0f };
        acc0 = __builtin_amdgcn_wmma_f32_16x16x4_f32(false, av, false, bv0, (short)0, acc0, false, false);
        acc1 = __builtin_amdgcn_wmma_f32_16x16x4_f32(false, av, false, bv1, (short)0, acc1, false, false);
        acc2 = __builtin_amdgcn_wmma_f32_16x16x4_f32(false, av, false, bv2, (short)0, acc2, false, false);
        acc3 = __builtin_amdgcn_wmma_f32_16x16x4_f32(false, av, false, bv3, (short)0, acc3, false, false);
    }

    // epilogue + guarded stores
    v8f accs[4] = { acc0, acc1, acc2, acc3 };
#pragma unroll
    for (int s = 0; s < 4; ++s) {
        const int n = nBase + s * 16 + nl;
        if (n >= N) continue;
        float bbv = 0.0f;
        if (EPI == 1 || EPI == 2) bbv = bias[n];
#pragma unroll
        for (int v = 0; v < 8; ++v) {
            const int row = mBase + v + 8 * hi;   // documented C/D layout
            if (row >= M) continue;
            float val = accs[s][v];
            if (EPI == 0) {
                val *= alpha;
            } else if (EPI == 1) {
                val += bbv;
            } else if (EPI == 2) {
                val += bbv;
                val = 0.5f * val * (1.0f + erff(val * 0.70710678118654752f));
            } else { // EPI == 3
                val *= rs[offR + row];
            }
            C[offC + (long long)row * ldc + n] = val;
        }
    }
}

// ---------------------------------------------------------------------------
// Small helper kernels
// ---------------------------------------------------------------------------

// patches (L,2,B,D) -> lin/lout (B,L,D)
__global__ void k_split(const float* __restrict__ patches,
                        float* __restrict__ lin, float* __restrict__ lout)
{
    long long i = (long long)blockIdx.x * blockDim.x + threadIdx.x;
    if (i >= cfg::BLD) return;
    int d = (int)(i % cfg::D);
    long long r = i / cfg::D;                 // b*L + l
    int b = (int)(r / cfg::L), l = (int)(r % cfg::L);
    const float* src = patches + (((long long)l * 2) * cfg::B + b) * cfg::D + d;
    lin[i]  = src[0];
    lout[i] = src[(long long)cfg::B * cfg::D];
}

// out = LayerNorm(x [+ y]) over D=243, one block per row
__global__ __launch_bounds__(256) void k_ln(const float* __restrict__ x,
                                            const float* __restrict__ y,
                                            const float* __restrict__ w,
                                            const float* __restrict__ bb,
                                            float* __restrict__ out)
{
    __shared__ float red[256];
    const long long row = blockIdx.x;
    const int t = threadIdx.x;
    float v = 0.0f;
    if (t < cfg::D) {
        v = x[row * cfg::D + t];
        if (y) v += y[row * cfg::D + t];
    }
    red[t] = v; __syncthreads();
    for (int s = 128; s; s >>= 1) { if (t < s) red[t] += red[t + s]; __syncthreads(); }
    const float mu = red[0] * (1.0f / cfg::D);
    __syncthreads();
    const float c = (t < cfg::D) ? (v - mu) : 0.0f;
    red[t] = c * c; __syncthreads();
    for (int s = 128; s; s >>= 1) { if (t < s) red[t] += red[t + s]; __syncthreads(); }
    const float var = red[0] * (1.0f / cfg::D);
    if (t < cfg::D)
        out[row * cfg::D + t] = c * rsqrtf(var + 1e-5f) * w[t] + bb[t];
}

// diag[h*BL + row] = 0.5*dn^2 * sum_f x[row, h*F+f]^2   (0.5*dn^2 = 1/18)
__global__ __launch_bounds__(256) void k_diag(const float* __restrict__ x,
                                              float* __restrict__ diag)
{
    __shared__ float red[256];
    const long long row = blockIdx.x;
    const int t = threadIdx.x;
    float v = (t < cfg::D) ? x[row * cfg::D + t] : 0.0f;
    red[t] = v * v; __syncthreads();
    if (t < cfg::H) {
        float s = 0.0f;
        for (int f = 0; f < cfg::F; ++f) s += red[t * cfg::F + f];
        diag[(long long)t * cfg::BL + row] = s * (1.0f / 18.0f);
    }
}

// rowmax[row] = max_j p[row, j], one block per p-row (H*BL rows)
__global__ __launch_bounds__(256) void k_rowmax(const float* __restrict__ p,
                                                float* __restrict__ rowmax)
{
    __shared__ float red[256];
    const long long row = blockIdx.x;
    const int t = threadIdx.x;
    red[t] = (t < cfg::NF) ? p[row * cfg::NF + t] : -3.402823466e38f;
    __syncthreads();
    for (int s = 128; s; s >>= 1) { if (t < s) red[t] = fmaxf(red[t], red[t + s]); __syncthreads(); }
    if (t == 0) rowmax[row] = red[0];
}

// kmax[hb] = max_n rowmax[hb*L + n]
__global__ __launch_bounds__(256) void k_colmax(const float* __restrict__ rowmax,
                                                float* __restrict__ kmax)
{
    __shared__ float red[256];
    const int z = blockIdx.x;
    const int t = threadIdx.x;
    float m = -3.402823466e38f;
    for (int i = t; i < cfg::L; i += 256) m = fmaxf(m, rowmax[(long long)z * cfg::L + i]);
    red[t] = m; __syncthreads();
    for (int s = 128; s; s >>= 1) { if (t < s) red[t] = fmaxf(red[t], red[t + s]); __syncthreads(); }
    if (t == 0) kmax[z] = red[0];
}

// p[i] = ratio * (exp(p[i] - diag[row] - m) + 1e-4);  m = rowmax (query) or kmax (key)
__global__ void k_exp(float* __restrict__ p, const float* __restrict__ diag,
                      const float* __restrict__ rowmax, const float* __restrict__ kmax,
                      int isQ)
{
    const long long tot = cfg::HBL * cfg::NF;
    long long i = (long long)blockIdx.x * blockDim.x + threadIdx.x;
    if (i >= tot) return;
    const long long row = i / cfg::NF;
    const long long hb  = row / cfg::L;
    const float m = isQ ? rowmax[row] : kmax[hb];
    p[i] = cfg::RATIO * (expf(p[i] - diag[row] - m) + 1e-4f);
}

// ksum[hb*NF + j] = sum_n pk[(hb*L+n)*NF + j]
__global__ __launch_bounds__(256) void k_ksum(const float* __restrict__ p,
                                              float* __restrict__ ksum)
{
    const int z = blockIdx.x;
    const int j = threadIdx.x;
    if (j >= cfg::NF) return;
    const float* base = p + (long long)z * cfg::L * cfg::NF + j;
    float s = 0.0f;
    for (int n = 0; n < cfg::L; ++n) s += base[(long long)n * cfg::NF];
    ksum[(long long)z * cfg::NF + j] = s;
}

// dinv[row] = 1 / sum_j pq[row,j] * ksum[hb,j]
__global__ __launch_bounds__(256) void k_dinv(const float* __restrict__ p,
                                              const float* __restrict__ ksum,
                                              float* __restrict__ dinv)
{
    __shared__ float red[256];
    const long long row = blockIdx.x;
    const long long hb  = row / cfg::L;
    const int t = threadIdx.x;
    red[t] = (t < cfg::NF) ? p[row * cfg::NF + t] * ksum[hb * cfg::NF + t] : 0.0f;
    __syncthreads();
    for (int s = 128; s; s >>= 1) { if (t < s) red[t] += red[t + s]; __syncthreads(); }
    if (t == 0) dinv[row] = 1.0f / red[0];
}

// ---------------------------------------------------------------------------
// Host-side orchestration
// ---------------------------------------------------------------------------
static inline void launch_gemm(hipStream_t st, int epi,
    const float* A, long long sAm, long long sAk, long long aLo, long long aHi,
    const float* Bm, long long sBk, long long sBn, long long bLo, long long bHi,
    float* C, long long ldc, long long cLo, long long cHi,
    int M, int N, int K, int nb, int bdiv, float alpha,
    const float* bias, const float* rs, long long rLo, long long rHi)
{
    dim3 g((unsigned)((N + 63) / 64), (unsigned)((M + 63) / 64), (unsigned)nb);
    dim3 t(128);
    switch (epi) {
    case 0: k_gemm<0><<<g, t, 0, st>>>(A,sAm,sAk,aLo,aHi, Bm,sBk,sBn,bLo,bHi, C,ldc,cLo,cHi, M,N,K,bdiv,alpha,bias,rs,rLo,rHi); break;
    case 1: k_gemm<1><<<g, t, 0, st>>>(A,sAm,sAk,aLo,aHi, Bm,sBk,sBn,bLo,bHi, C,ldc,cLo,cHi, M,N,K,bdiv,alpha,bias,rs,rLo,rHi); break;
    case 2: k_gemm<2><<<g, t, 0, st>>>(A,sAm,sAk,aLo,aHi, Bm,sBk,sBn,bLo,bHi, C,ldc,cLo,cHi, M,N,K,bdiv,alpha,bias,rs,rLo,rHi); break;
    default:k_gemm<3><<<g, t, 0, st>>>(A,sAm,sAk,aLo,aHi, Bm,sBk,sBn,bLo,bHi, C,ldc,cLo,cHi, M,N,K,bdiv,alpha,bias,rs,rLo,rHi); break;
    }
}

// Performer fast_attention; x = q = k in (B,L,D), v in (B,L,D), out (B,L,D)
static void attention(hipStream_t st, const float* x, const float* v, const float* proj,
                      float* out, float* p, float* ctx, float* ksum,
                      float* diag, float* rowmax, float* kmax, float* dinv)
{
    using namespace cfg;
    const int HB = H * B;
    // xd = dn * (x_head @ proj^T), batched over heads -> p[h, b, n, j]
    auto xd = [&]() {
        launch_gemm(st, 0, x, D, 1, 0, F,
                    proj, 1, F, 0, 0,
                    p, NF, 0, BL * NF,
                    (int)BL, NF, F, H, 1, DN, nullptr, nullptr, 0, 0);
    };
    xd();
    k_diag<<<(unsigned)BL, 256, 0, st>>>(x, diag);
    k_rowmax<<<(unsigned)HBL, 256, 0, st>>>(p, rowmax);
    k_colmax<<<HB, 256, 0, st>>>(rowmax, kmax);
    // key feature map (global max), then ksum and ctx = pk^T @ v
    const long long tot = HBL * NF;
    k_exp<<<(unsigned)((tot + 255) / 256), 256, 0, st>>>(p, diag, rowmax, kmax, 0);
    k_ksum<<<HB, 256, 0, st>>>(p, ksum);
    launch_gemm(st, 0, p, 1, NF, (long long)L * NF, (long long)B * L * NF,
                v, D, 1, (long long)L * D, F,
                ctx, F, (long long)NF * F, (long long)B * NF * F,
                NF, F, L, HB, B, 1.0f, nullptr, nullptr, 0, 0);
    // query feature map (recompute xd: cheaper than a second 96MB buffer)
    xd();
    k_exp<<<(unsigned)((tot + 255) / 256), 256, 0, st>>>(p, diag, rowmax, kmax, 1);
    k_dinv<<<(unsigned)HBL, 256, 0, st>>>(p, ksum, dinv);
    // out[b,n,h*F+e] = dinv * (pq @ ctx)
    launch_gemm(st, 3, p, NF, 1, (long long)L * NF, (long long)B * L * NF,
                ctx, F, 1, (long long)NF * F, (long long)B * NF * F,
                out, D, (long long)L * D, F,
                L, F, NF, HB, B, 1.0f, nullptr, dinv, L, BL);
}

static void feed_forward(hipStream_t st, const float* x,
                         const float* w1, const float* b1,
                         const float* w2, const float* b2,
                         float* hid, float* out)
{
    using namespace cfg;
    launch_gemm(st, 2, x, D, 1, 0, 0, w1, FFH, 1, 0, 0,
                hid, FFH, 0, 0, (int)BL, FFH, D, 1, 1, 1.0f, b1, nullptr, 0, 0);
    launch_gemm(st, 1, hid, FFH, 1, 0, 0, w2, D, 1, 0, 0,
                out, D, 0, 0, (int)BL, D, FFH, 1, 1, 1.0f, b2, nullptr, 0, 0);
}

extern "C" void kernel_launch(void* const* d_in, const int* in_sizes, int n_in,
                              void* d_out, int out_size, void* d_ws, size_t ws_size,
                              hipStream_t stream)
{
    using namespace cfg;
    const float* patches   = (const float*)d_in[0];
    const float* ln_w      = (const float*)d_in[1];
    const float* ln_b      = (const float*)d_in[2];
    const float* enc_proj  = (const float*)d_in[3];
    const float* enc_w1    = (const float*)d_in[4];
    const float* enc_b1    = (const float*)d_in[5];
    const float* enc_w2    = (const float*)d_in[6];
    const float* enc_b2    = (const float*)d_in[7];
    const float* dec1_proj = (const float*)d_in[8];
    const float* dec2_proj = (const float*)d_in[9];
    const float* dec_w1    = (const float*)d_in[10];
    const float* dec_b1    = (const float*)d_in[11];
    const float* dec_w2    = (const float*)d_in[12];
    const float* dec_b2    = (const float*)d_in[13];
    (void)in_sizes; (void)n_in; (void)out_size;

    // workspace arena (floats)
    const long long NEED = 4 * BLD + PEL + CTXE + KSUME + 3 * HBL + 128;
    if (ws_size < (size_t)NEED * sizeof(float)) return;  // insufficient scratch
    float* ws     = (float*)d_ws;
    float* lin    = ws;                 // (B,L,D) encoder state
    float* lout   = lin + BLD;          // (B,L,D) decoder state
    float* t0     = lout + BLD;         // attention / ff output
    float* t1     = t0 + BLD;           // ff_in / a1
    float* p      = t1 + BLD;           // (H,B,L,NF) feature maps, aliases FF hidden
    float* ctx    = p + PEL;            // (H*B, NF, F)
    float* ksum   = ctx + CTXE;         // (H*B, NF)
    float* diag   = ksum + KSUME;       // (H*BL)
    float* rowmax = diag + HBL;         // (H*BL)
    float* dinv   = rowmax + HBL;       // (H*BL)
    float* kmax   = dinv + HBL;         // (H*B)

    k_split<<<(unsigned)((BLD + 255) / 256), 256, 0, stream>>>(patches, lin, lout);

    // encoder
    for (int i = 0; i < 2; ++i) {
        attention(stream, lin, lin, enc_proj + (long long)i * NF * F,
                  t0, p, ctx, ksum, diag, rowmax, kmax, dinv);
        k_ln<<<(unsigned)BL, 256, 0, stream>>>(t0, lin, ln_w, ln_b, t1);     // ff_in
        feed_forward(stream, t1, enc_w1 + (long long)i * D * FFH, enc_b1 + (long long)i * FFH,
                     enc_w2 + (long long)i * FFH * D, enc_b2 + (long long)i * D, p, t0);
        k_ln<<<(unsigned)BL, 256, 0, stream>>>(t0, t1, ln_w, ln_b, lin);     // new l_in
    }

    // decoder
    float* outf = (float*)d_out;
    for (int i = 0; i < 2; ++i) {
        attention(stream, lout, lout, dec1_proj + (long long)i * NF * F,
                  t0, p, ctx, ksum, diag, rowmax, kmax, dinv);
        k_ln<<<(unsigned)BL, 256, 0, stream>>>(t0, lout, ln_w, ln_b, t1);    // a1
        attention(stream, lin, t1, dec2_proj + (long long)i * NF * F,
                  t0, p, ctx, ksum, diag, rowmax, kmax, dinv);
        k_ln<<<(unsigned)BL, 256, 0, stream>>>(t0, t1, ln_w, ln_b, lout);    // a2
        feed_forward(stream, lout, dec_w1 + (long long)i * D * FFH, dec_b1 + (long long)i * FFH,
                     dec_w2 + (long long)i * FFH * D, dec_b2 + (long long)i * D, p, t0);
        float* dst = (i == 1) ? outf : lout;                                 // final layer -> d_out
        k_ln<<<(unsigned)BL, 256, 0, stream>>>(t0, lout, ln_w, ln_b, dst);   // new l_out
    }
}